// ChebConv_47055661695424
// MI455X (gfx1250) — compile-verified
//
#include <hip/hip_runtime.h>

#define LAM 2.0f

typedef __attribute__((ext_vector_type(2))) float v2f;
typedef __attribute__((ext_vector_type(8))) float v8f;

// ---------------------------------------------------------------------------
// Stage 0: zero deg + h scatter buffer (ws is poisoned to 0xAA by harness)
// ---------------------------------------------------------------------------
__global__ void zero_kernel(float* __restrict__ p, long long n) {
    long long i = (long long)blockIdx.x * blockDim.x + threadIdx.x;
    if (i < n) p[i] = 0.0f;
}

// ---------------------------------------------------------------------------
// Stage 1: degree of each destination node (fp32 atomic, L2-resident)
// ---------------------------------------------------------------------------
__global__ void degree_kernel(const int* __restrict__ dst, float* __restrict__ deg, int E) {
    int i = blockIdx.x * blockDim.x + threadIdx.x;
    if (i < E) atomicAdd(&deg[dst[i]], 1.0f);
}

// ---------------------------------------------------------------------------
// Stage 2: norm[i] = max(deg,1)^-0.5  (in place over deg buffer)
// ---------------------------------------------------------------------------
__global__ void norm_kernel(float* __restrict__ deg, int N) {
    int i = blockIdx.x * blockDim.x + threadIdx.x;
    if (i < N) deg[i] = 1.0f / sqrtf(fmaxf(deg[i], 1.0f));
}

// ---------------------------------------------------------------------------
// Stage 3/5: sparse aggregation scatter: h[dst,f] += x[src,f]*norm[src]
// One thread per (edge, feature): coalesced 256B row gathers, clustered
// atomics into the 25.6MB h buffer which stays resident in the 192MB L2.
// ---------------------------------------------------------------------------
__global__ void scatter_kernel(const float* __restrict__ x, const float* __restrict__ norm,
                               const int* __restrict__ src, const int* __restrict__ dst,
                               float* __restrict__ h, long long total) {
    long long i = (long long)blockIdx.x * blockDim.x + threadIdx.x;
    if (i >= total) return;
    int e = (int)(i >> 6);
    int f = (int)(i & 63);
    int s = src[e];
    int d = dst[e];
    float v = x[(long long)s * 64 + f] * norm[s];
    atomicAdd(&h[(long long)d * 64 + f], v);
}

// ---------------------------------------------------------------------------
// Stage 4: Tx1 = (-2/lam)*h*norm[node] + (2/lam-1)*feat ; reset h for pass 2
// ---------------------------------------------------------------------------
__global__ void cheb1_kernel(const float* __restrict__ feat, const float* __restrict__ norm,
                             float* __restrict__ h, float* __restrict__ Tx1, long long total) {
    long long i = (long long)blockIdx.x * blockDim.x + threadIdx.x;
    if (i >= total) return;
    const float ch = -2.0f / LAM;
    const float cx = 2.0f / LAM - 1.0f;
    int node = (int)(i >> 6);
    Tx1[i] = ch * h[i] * norm[node] + cx * feat[i];
    h[i] = 0.0f;
}

// ---------------------------------------------------------------------------
// Stage 6: fused Cheb-2 recurrence + triple GEMM via V_WMMA_F32_16X16X4_F32.
//   out[n,:] = feat@W0 + Tx1@W1 + Tx2@W2 + bias
//   Tx2 computed on the fly: (-4/lam)*h2*norm + (4/lam-2)*Tx1 - feat
// One wave per 16x16 output tile; K=64 in 16 steps of 4; 48 WMMA per wave.
// EXEC is all-1s (full 256-thread blocks, no divergence) as WMMA requires.
// ---------------------------------------------------------------------------
__global__ __launch_bounds__(256) void cheb_gemm_wmma(
    const float* __restrict__ feat, const float* __restrict__ Tx1,
    const float* __restrict__ h2,   const float* __restrict__ norm,
    const float* __restrict__ W,    const float* __restrict__ bias,
    float* __restrict__ out) {
    const int lane = threadIdx.x & 31;
    const int wave = threadIdx.x >> 5;
    const int gw   = blockIdx.x * 8 + wave;
    const int row_tile = gw >> 2;   // 6250 row tiles of 16 nodes
    const int n_tile   = gw & 3;    // 4 column tiles of 16
    const int l15  = lane & 15;
    const int half = lane >> 4;     // 0: K pair {0,1}, 1: K pair {2,3}
    const int koff = half * 2;

    const int arow = row_tile * 16 + l15;   // A-matrix row this lane feeds
    const int ncol = n_tile * 16 + l15;     // B/D column this lane feeds

    const float nrm = norm[arow];
    const float* fr = feat + (long long)arow * 64;
    const float* tr = Tx1  + (long long)arow * 64;
    const float* hr = h2   + (long long)arow * 64;
    const float* W0 = W;
    const float* W1 = W + 4096;
    const float* W2 = W + 8192;

    const float ch2 = -4.0f / LAM;
    const float cx2 =  4.0f / LAM - 2.0f;

    v8f acc = {};
#pragma unroll
    for (int k = 0; k < 64; k += 4) {
        const int ka = k + koff;
        // A operands: lane holds A[m][ka], A[m][ka+1]
        float f0 = fr[ka],  f1 = fr[ka + 1];
        float t0 = tr[ka],  t1 = tr[ka + 1];
        float g0 = hr[ka],  g1 = hr[ka + 1];
        v2f a0 = {f0, f1};
        v2f a1 = {t0, t1};
        v2f a2 = {ch2 * g0 * nrm + cx2 * t0 - f0,
                  ch2 * g1 * nrm + cx2 * t1 - f1};
        // B operands: lane holds B[ka][n], B[ka+1][n]
        v2f b0 = {W0[ka * 64 + ncol], W0[(ka + 1) * 64 + ncol]};
        v2f b1 = {W1[ka * 64 + ncol], W1[(ka + 1) * 64 + ncol]};
        v2f b2 = {W2[ka * 64 + ncol], W2[(ka + 1) * 64 + ncol]};
        acc = __builtin_amdgcn_wmma_f32_16x16x4_f32(false, a0, false, b0,
                                                    (short)0, acc, false, false);
        acc = __builtin_amdgcn_wmma_f32_16x16x4_f32(false, a1, false, b1,
                                                    (short)0, acc, false, false);
        acc = __builtin_amdgcn_wmma_f32_16x16x4_f32(false, a2, false, b2,
                                                    (short)0, acc, false, false);
    }

    const float b = bias[ncol];
    // D layout: vgpr r -> row (r + 8*half), column = ncol
#pragma unroll
    for (int r = 0; r < 8; ++r) {
        int orow = row_tile * 16 + r + 8 * half;
        out[(long long)orow * 64 + ncol] = acc[r] + b;
    }
}

// ---------------------------------------------------------------------------
// Host-side orchestration (graph-capture safe: stream-only launches)
// Inputs: [0]=feat (N*64 f32), [1]=W (3*64*64 f32), [2]=bias (64 f32),
//         [3]=edge_src (E i32), [4]=edge_dst (E i32)
// Workspace: [norm: N][h: N*64][Tx1: N*64] floats (~51.6 MB)
// ---------------------------------------------------------------------------
extern "C" void kernel_launch(void* const* d_in, const int* in_sizes, int n_in,
                              void* d_out, int out_size, void* d_ws, size_t ws_size,
                              hipStream_t stream) {
    const float* feat = (const float*)d_in[0];
    const float* W    = (const float*)d_in[1];
    const float* bias = (const float*)d_in[2];
    const int*   esrc = (const int*)d_in[3];
    const int*   edst = (const int*)d_in[4];
    float* out = (float*)d_out;

    const int N = in_sizes[0] / 64;   // 100000 (divisible by 16)
    const int E = in_sizes[3];        // 1600000

    float* ws   = (float*)d_ws;
    float* norm = ws;                               // N floats (deg -> norm)
    float* h    = ws + N;                           // N*64 floats
    float* Tx1  = ws + N + (long long)N * 64;       // N*64 floats

    const long long nf  = (long long)N * 64;
    const long long tot = (long long)E * 64;
    const long long zn  = (long long)N * 65;        // norm + h contiguous

    zero_kernel<<<(unsigned)((zn + 255) / 256), 256, 0, stream>>>(ws, zn);
    degree_kernel<<<(E + 255) / 256, 256, 0, stream>>>(edst, norm, E);
    norm_kernel<<<(N + 255) / 256, 256, 0, stream>>>(norm, N);

    // h = scatter(feat*norm)
    scatter_kernel<<<(unsigned)((tot + 255) / 256), 256, 0, stream>>>(
        feat, norm, esrc, edst, h, tot);
    // Tx1 = -2/lam * h*norm + (2/lam-1)*feat ; h = 0
    cheb1_kernel<<<(unsigned)((nf + 255) / 256), 256, 0, stream>>>(
        feat, norm, h, Tx1, nf);
    // h = scatter(Tx1*norm)
    scatter_kernel<<<(unsigned)((tot + 255) / 256), 256, 0, stream>>>(
        Tx1, norm, esrc, edst, h, tot);

    // out = feat@W0 + Tx1@W1 + Tx2(h,Tx1,feat)@W2 + bias
    const int waves = (N / 16) * 4;                 // 25000 tiles
    cheb_gemm_wmma<<<waves / 8, 256, 0, stream>>>(feat, Tx1, h, norm, W, bias, out);
}